// WaveNetModel_80693845557932
// MI455X (gfx1250) — compile-verified
//
#include <hip/hip_runtime.h>
#include <cstddef>

// ---------------- problem constants (from reference) ----------------
#define MU_   256
#define NR_   32
#define NS_   512
#define B_    4
#define T_    6144
#define TOUT_ 1029
#define NLAY_ 50
#define TT    64     // time tile per block in the layer kernel

typedef __attribute__((ext_vector_type(2))) float v2f;
typedef __attribute__((ext_vector_type(8))) float v8f;

// fp32 WMMA: D(16x16,f32) = A(16x4,f32) x B(4x16,f32) + C
__device__ __forceinline__ v8f wmma_f32_k4(v2f a, v2f b, v8f c) {
  return __builtin_amdgcn_wmma_f32_16x16x4_f32(
      /*neg_a=*/false, a, /*neg_b=*/false, b,
      /*c_mod=*/(short)0, c, /*reuse_a=*/false, /*reuse_b=*/false);
}

// ---------------- embedding: x0[b][c][t] = W_in[c][inp[b][t]] + b_in[c] ----
__global__ __launch_bounds__(256) void wavenet_embed(
    const int* __restrict__ inp, const float* __restrict__ W_in,
    const float* __restrict__ b_in, float* __restrict__ x0) {
  int idx = blockIdx.x * 256 + threadIdx.x;     // over B*T
  if (idx >= B_ * T_) return;
  int b = idx / T_, t = idx - b * T_;
  int m = inp[idx];
#pragma unroll
  for (int c = 0; c < NR_; ++c)
    x0[((size_t)b * NR_ + c) * T_ + t] = W_in[c * MU_ + m] + b_in[c];
}

// ---------------- repack W_skip [L][NS][NR] -> [NS][L*NR]; sum b_skip ------
__global__ __launch_bounds__(256) void repack_skip(
    const float* __restrict__ W_skip, const float* __restrict__ b_skip,
    float* __restrict__ Wp, float* __restrict__ bp) {
  int idx = blockIdx.x * 256 + threadIdx.x;
  if (idx < NS_) {
    float s = 0.f;
    for (int l = 0; l < NLAY_; ++l) s += b_skip[l * NS_ + idx];
    bp[idx] = s;
  }
  if (idx >= NS_ * (NLAY_ * NR_)) return;
  int o = idx / (NLAY_ * NR_), k = idx - o * (NLAY_ * NR_);
  int l = k >> 5, i = k & 31;
  Wp[idx] = W_skip[((size_t)l * NS_ + o) * NR_ + i];
}

// ---------------- one WaveNet layer ---------------------------------------
// x_in [B][32][T_in] -> x_out [B][32][T_in-d]; gate (cropped to last TOUT
// cols) appended to g_buf at K-row l*32+c, layout [(l*32+i)][b][t].
__global__ __launch_bounds__(256) void wavenet_layer(
    const float* __restrict__ x_in, float* __restrict__ x_out,
    float* __restrict__ g_buf,
    const float* __restrict__ W_sig,  const float* __restrict__ b_sig,
    const float* __restrict__ W_tanh, const float* __restrict__ b_tanh,
    const float* __restrict__ W_res,  const float* __restrict__ b_res,
    int l, int d, int T_in) {
  const int T_out = T_in - d;
  const int b  = blockIdx.y;
  const int t0 = blockIdx.x * TT;
  if (t0 >= T_out) return;
  const int cols = min(TT, T_out - t0);
  const int tid  = threadIdx.x;
  const int wave = tid >> 5, lane = tid & 31;
  const int half = lane >> 4, lm = lane & 15;

  __shared__ float sWs0[NR_ * 33], sWs1[NR_ * 33];
  __shared__ float sWt0[NR_ * 33], sWt1[NR_ * 33];
  __shared__ float sWr [NR_ * 33];
  __shared__ float sX0[NR_ * TT], sX1[NR_ * TT], sG[NR_ * TT];

  // stage layer weights (32x32 each; tap-split for the dilated convs)
  for (int idx = tid; idx < NR_ * NR_; idx += 256) {
    int o = idx >> 5, i = idx & 31;
    const float* ws = W_sig  + (size_t)l * NR_ * NR_ * 2 + (size_t)o * (NR_ * 2) + i * 2;
    const float* wt = W_tanh + (size_t)l * NR_ * NR_ * 2 + (size_t)o * (NR_ * 2) + i * 2;
    sWs0[o * 33 + i] = ws[0];  sWs1[o * 33 + i] = ws[1];
    sWt0[o * 33 + i] = wt[0];  sWt1[o * 33 + i] = wt[1];
    sWr [o * 33 + i] = W_res[(size_t)l * NR_ * NR_ + (size_t)o * NR_ + i];
  }
  // stage both conv taps of x
  const float* xb = x_in + (size_t)b * NR_ * T_in;
  for (int idx = tid; idx < NR_ * TT; idx += 256) {
    int c = idx >> 6, t = idx & (TT - 1);
    float v0 = 0.f, v1 = 0.f;
    if (t < cols) {
      v0 = xb[(size_t)c * T_in + t0 + t];
      v1 = xb[(size_t)c * T_in + t0 + t + d];
    }
    sX0[c * TT + t] = v0;
    sX1[c * TT + t] = v1;
  }
  __syncthreads();

  // 8 waves cover the 2(M) x 4(N) = 8 output tiles of the 32 x 64 result
  const int M0 = (wave & 1) * 16;
  const int N0 = (wave >> 1) * 16;

  v8f cs = {}, ct = {};
#pragma unroll
  for (int r = 0; r < 8; ++r) {
    int row = M0 + r + 8 * half;
    cs[r] = b_sig [l * NR_ + row];
    ct[r] = b_tanh[l * NR_ + row];
  }
#pragma unroll
  for (int k0 = 0; k0 < NR_; k0 += 4) {
    int ka = k0 + 2 * half;
    v2f b0, b1, a;
    b0.x = sX0[(ka + 0) * TT + N0 + lm];  b0.y = sX0[(ka + 1) * TT + N0 + lm];
    b1.x = sX1[(ka + 0) * TT + N0 + lm];  b1.y = sX1[(ka + 1) * TT + N0 + lm];
    a.x = sWs0[(M0 + lm) * 33 + ka];  a.y = sWs0[(M0 + lm) * 33 + ka + 1];
    cs = wmma_f32_k4(a, b0, cs);
    a.x = sWs1[(M0 + lm) * 33 + ka];  a.y = sWs1[(M0 + lm) * 33 + ka + 1];
    cs = wmma_f32_k4(a, b1, cs);
    a.x = sWt0[(M0 + lm) * 33 + ka];  a.y = sWt0[(M0 + lm) * 33 + ka + 1];
    ct = wmma_f32_k4(a, b0, ct);
    a.x = sWt1[(M0 + lm) * 33 + ka];  a.y = sWt1[(M0 + lm) * 33 + ka + 1];
    ct = wmma_f32_k4(a, b1, ct);
  }
  // gate = sigmoid(cs) * tanh(ct)
#pragma unroll
  for (int r = 0; r < 8; ++r) {
    float sg = 1.f / (1.f + __expf(-cs[r]));
    float th = tanhf(ct[r]);
    sG[(M0 + r + 8 * half) * TT + N0 + lm] = sg * th;
  }
  __syncthreads();

  // write cropped gate slice into the big skip-GEMM K buffer
  const int crop = T_out - TOUT_;
  for (int idx = tid; idx < NR_ * TT; idx += 256) {
    int c = idx >> 6, t = idx & (TT - 1);
    int tg = t0 + t;
    if (t < cols && tg >= crop)
      g_buf[(((size_t)l * NR_ + c) * B_ + b) * TOUT_ + (tg - crop)] = sG[c * TT + t];
  }

  // x_new = W_res @ gate + b_res + x[:, d:]
  v8f cr = {};
#pragma unroll
  for (int r = 0; r < 8; ++r) cr[r] = b_res[l * NR_ + M0 + r + 8 * half];
#pragma unroll
  for (int k0 = 0; k0 < NR_; k0 += 4) {
    int ka = k0 + 2 * half;
    v2f a, bb;
    a.x  = sWr[(M0 + lm) * 33 + ka];  a.y  = sWr[(M0 + lm) * 33 + ka + 1];
    bb.x = sG[(ka + 0) * TT + N0 + lm];  bb.y = sG[(ka + 1) * TT + N0 + lm];
    cr = wmma_f32_k4(a, bb, cr);
  }
  if (N0 + lm < cols) {
    float* xo = x_out + (size_t)b * NR_ * T_out;
#pragma unroll
    for (int r = 0; r < 8; ++r) {
      int row = M0 + r + 8 * half;
      xo[(size_t)row * T_out + t0 + N0 + lm] = cr[r] + sX1[row * TT + N0 + lm];
    }
  }
}

// ---------------- generic GEMM: Y[b][m][n] = bias[m] + W[m][k] f(X) -------
// X row k for batch b lives at X + k*xrs + b*xbo + n ; f = ELU if act.
__global__ __launch_bounds__(256) void gemm_act(
    const float* __restrict__ W, const float* __restrict__ X,
    const float* __restrict__ bias, float* __restrict__ Y,
    int M, int K, int N, long xrs, long xbo, int act) {
  const int b  = blockIdx.z;
  const int Mg = blockIdx.y * 64;
  const int Ng = blockIdx.x * 64;
  const int tid = threadIdx.x, wave = tid >> 5, lane = tid & 31;
  const int half = lane >> 4, lm = lane & 15;
  __shared__ float sW[64 * 33];   // [m][k] chunk, Kc=32, padded
  __shared__ float sX[32 * 65];   // [k][n] chunk, padded

  const int mt0 = wave >> 2, nt0 = wave & 3;  // tile ids wave, wave+8
  const int mt1 = mt0 + 2;
  v8f c0 = {}, c1 = {};

  for (int k0g = 0; k0g < K; k0g += 32) {
    __syncthreads();
    for (int idx = tid; idx < 64 * 32; idx += 256) {
      int m = idx >> 5, k = idx & 31;
      sW[m * 33 + k] = W[(size_t)(Mg + m) * K + k0g + k];
    }
    for (int idx = tid; idx < 32 * 64; idx += 256) {
      int k = idx >> 6, n = idx & 63;
      float v = 0.f;
      if (Ng + n < N) v = X[(size_t)(k0g + k) * xrs + (size_t)b * xbo + Ng + n];
      if (act) v = v > 0.f ? v : (__expf(v) - 1.f);   // ELU(alpha=1)
      sX[k * 65 + n] = v;
    }
    __syncthreads();
#pragma unroll
    for (int kk = 0; kk < 32; kk += 4) {
      int ka = kk + 2 * half;
      v2f a0, a1, bb;
      a0.x = sW[(mt0 * 16 + lm) * 33 + ka];  a0.y = sW[(mt0 * 16 + lm) * 33 + ka + 1];
      a1.x = sW[(mt1 * 16 + lm) * 33 + ka];  a1.y = sW[(mt1 * 16 + lm) * 33 + ka + 1];
      bb.x = sX[(ka + 0) * 65 + nt0 * 16 + lm];
      bb.y = sX[(ka + 1) * 65 + nt0 * 16 + lm];
      c0 = wmma_f32_k4(a0, bb, c0);
      c1 = wmma_f32_k4(a1, bb, c1);
    }
  }
  int n = Ng + nt0 * 16 + lm;
  if (n < N) {
#pragma unroll
    for (int r = 0; r < 8; ++r) {
      int m = Mg + mt0 * 16 + r + 8 * half;
      Y[((size_t)b * M + m) * N + n] = c0[r] + bias[m];
    }
#pragma unroll
    for (int r = 0; r < 8; ++r) {
      int m = Mg + mt1 * 16 + r + 8 * half;
      Y[((size_t)b * M + m) * N + n] = c1[r] + bias[m];
    }
  }
}

// ---------------- host-side orchestration ---------------------------------
extern "C" void kernel_launch(void* const* d_in, const int* in_sizes, int n_in,
                              void* d_out, int out_size, void* d_ws, size_t ws_size,
                              hipStream_t stream) {
  const int*   inp    = (const int*)  d_in[0];
  const float* W_in   = (const float*)d_in[1];
  const float* b_in   = (const float*)d_in[2];
  const float* W_sig  = (const float*)d_in[3];
  const float* b_sig  = (const float*)d_in[4];
  const float* W_tanh = (const float*)d_in[5];
  const float* b_tanh = (const float*)d_in[6];
  const float* W_skip = (const float*)d_in[7];
  const float* b_skip = (const float*)d_in[8];
  const float* W_res  = (const float*)d_in[9];
  const float* b_res  = (const float*)d_in[10];
  const float* W_p1   = (const float*)d_in[11];
  const float* b_p1   = (const float*)d_in[12];
  const float* W_p2   = (const float*)d_in[13];
  const float* b_p2   = (const float*)d_in[14];

  float* ws    = (float*)d_ws;
  float* x0    = ws;                                        // B*32*T
  float* x1    = x0 + (size_t)B_ * NR_ * T_;                // B*32*T
  float* g_buf = x1 + (size_t)B_ * NR_ * T_;                // L*32*B*TOUT
  float* Wp    = g_buf + (size_t)NLAY_ * NR_ * B_ * TOUT_;  // 512*1600
  float* bp    = Wp + (size_t)NS_ * NLAY_ * NR_;            // 512
  float* s     = bp + NS_;                                  // B*512*TOUT
  float* h2    = s + (size_t)B_ * NS_ * TOUT_;              // B*512*TOUT
  float* out   = (float*)d_out;                             // B*256*TOUT

  wavenet_embed<<<(B_ * T_ + 255) / 256, 256, 0, stream>>>(inp, W_in, b_in, x0);
  repack_skip<<<(NS_ * NLAY_ * NR_ + 255) / 256, 256, 0, stream>>>(W_skip, b_skip, Wp, bp);

  int T_cur = T_;
  float* xa = x0; float* xb = x1;
  int l = 0;
  for (int rep = 0; rep < 5; ++rep) {
    for (int i = 0; i < 10; ++i, ++l) {
      int d = 1 << i;
      int T_out = T_cur - d;
      dim3 g((T_out + TT - 1) / TT, B_);
      wavenet_layer<<<g, 256, 0, stream>>>(xa, xb, g_buf, W_sig, b_sig,
                                           W_tanh, b_tanh, W_res, b_res,
                                           l, d, T_cur);
      float* tmp = xa; xa = xb; xb = tmp;
      T_cur = T_out;
    }
  }

  {  // skip path as ONE GEMM over K = 50*32 = 1600
    dim3 g((TOUT_ + 63) / 64, NS_ / 64, B_);
    gemm_act<<<g, 256, 0, stream>>>(Wp, g_buf, bp, s, NS_, NLAY_ * NR_, TOUT_,
                                    (long)B_ * TOUT_, (long)TOUT_, 0);
  }
  {  // h2 = W_p1 @ elu(s) + b_p1
    dim3 g((TOUT_ + 63) / 64, NS_ / 64, B_);
    gemm_act<<<g, 256, 0, stream>>>(W_p1, s, b_p1, h2, NS_, NS_, TOUT_,
                                    (long)TOUT_, (long)NS_ * TOUT_, 1);
  }
  {  // out = W_p2 @ elu(h2) + b_p2
    dim3 g((TOUT_ + 63) / 64, MU_ / 64, B_);
    gemm_act<<<g, 256, 0, stream>>>(W_p2, h2, b_p2, out, MU_, NS_, TOUT_,
                                    (long)TOUT_, (long)NS_ * TOUT_, 1);
  }
}